// RGCN_1769526526087
// MI455X (gfx1250) — compile-verified
//
#include <hip/hip_runtime.h>

// ---------------------------------------------------------------------------
// RGCN (2-layer HeteroGraphConv, 9 relations) for MI455X / gfx1250 (wave32).
//
// Layer 1 (aggregate-first):  pre1 += (A_r_norm @ x) @ W1[r]   (K=128)
// Layer 2 (weight-first):     out  += A_r_norm @ (h1 @ W2[r])  (scatter 128)
// norm='both' folded into a per-edge scale dinv_src[s]*dinv_dst[d].
// GEMMs use V_WMMA_F32_16X16X4_F32 (f32 in, f32 acc), one wave = 16x64 tile
// (4 accumulators): per K-step, 1 A-fragment load + 4 B fragments loaded as
// a clause, then 4 back-to-back WMMAs.
// ---------------------------------------------------------------------------

#define R_REL 9
#define N_NODES 50000
#define E_EDGES 250000
#define F_IN 128
#define HID 256
#define OUT_F 128

typedef float v2f __attribute__((ext_vector_type(2)));
typedef float v8f __attribute__((ext_vector_type(8)));

// ---------------- degree computation --------------------------------------
__global__ void deg_count_kernel(const int* __restrict__ src,
                                 const int* __restrict__ dst,
                                 float* __restrict__ deg_src,
                                 float* __restrict__ deg_dst) {
    int i = blockIdx.x * blockDim.x + threadIdx.x;   // over R*E
    if (i >= R_REL * E_EDGES) return;
    int r = i / E_EDGES;
    atomicAdd(&deg_src[r * N_NODES + src[i]], 1.0f);
    atomicAdd(&deg_dst[r * N_NODES + dst[i]], 1.0f);
}

__global__ void deg_finalize_kernel(float* __restrict__ d, int n) {
    int i = blockIdx.x * blockDim.x + threadIdx.x;
    if (i >= n) return;
    d[i] = rsqrtf(fmaxf(d[i], 1.0f));                // max(deg,1)^-1/2
}

// ---------------- edge gather/scale/scatter (128 features, 1 wave/edge) ----
__global__ void edge_scatter128_kernel(const float* __restrict__ feat,  // [N,128]
                                       const int* __restrict__ src,     // [E]
                                       const int* __restrict__ dst,     // [E]
                                       const float* __restrict__ dis_s, // [N]
                                       const float* __restrict__ dis_d, // [N]
                                       float* __restrict__ out,         // [N,128]
                                       int nEdges, int useNorm) {
    int e    = (blockIdx.x * blockDim.x + threadIdx.x) >> 5;  // wave id = edge
    int lane = threadIdx.x & 31;
    if (e >= nEdges) return;
    int s = src[e];
    int d = dst[e];
    float sc = useNorm ? (dis_s[s] * dis_d[d]) : 1.0f;
    const float4 v = *(const float4*)(feat + (long)s * 128 + lane * 4);
    float* o = out + (long)d * 128 + lane * 4;
    atomicAdd(o + 0, v.x * sc);
    atomicAdd(o + 1, v.y * sc);
    atomicAdd(o + 2, v.z * sc);
    atomicAdd(o + 3, v.w * sc);
}

// ---------------- WMMA f32 GEMM: C[M,Nc] (+)= A[M,K] @ B[K,Nc] -------------
// One wave computes a 16x64 C strip = 4 x (16x16) tiles, sharing the A
// fragment across 4 WMMAs per K-step. Wave-uniform guard keeps EXEC all-ones
// (WMMA requirement). Tiles are assigned column-major so consecutive waves
// reuse the same B columns through cache.
// Fragment layouts per CDNA5 ISA §7.12.2:
//   A 16x4 : lane l -> M = l%16, VGPR j holds K = k0 + j + 2*(l/16)  (float2)
//   B 4x16 : lane l -> N = l%16, VGPR j holds K = k0 + j + 2*(l/16)
//   C 16x16: lane l -> N = l%16, VGPR j holds M = j + 8*(l/16)
__global__ void gemm_wmma_f32_kernel(const float* __restrict__ A,
                                     const float* __restrict__ B,
                                     float* __restrict__ C,
                                     int M, int Nc, int K, int accumulate) {
    int wavesPerBlock = blockDim.x >> 5;
    int w    = blockIdx.x * wavesPerBlock + (threadIdx.x >> 5);
    int lane = threadIdx.x & 31;
    int tilesN = Nc >> 6;                             // 64-wide strips
    int tilesM = M >> 4;
    if (w >= tilesM * tilesN) return;                 // wave-uniform exit
    int tn = w / tilesM;                              // column-major mapping
    int tm = w - tn * tilesM;
    int m0 = tm << 4, n0 = tn << 6;
    int row  = lane & 15;
    int half = lane >> 4;

    v8f acc[4];
#pragma unroll
    for (int t = 0; t < 4; ++t)
#pragma unroll
        for (int j = 0; j < 8; ++j) acc[t][j] = 0.0f;

    if (accumulate) {
#pragma unroll
        for (int t = 0; t < 4; ++t)
#pragma unroll
            for (int j = 0; j < 8; ++j)
                acc[t][j] = C[(long)(m0 + j + 8 * half) * Nc + n0 + 16 * t + row];
    }

    const float* Arow = A + (long)(m0 + row) * K + 2 * half;
    const float* Bp   = B + (long)(2 * half) * Nc + n0 + row;  // K-row pair base
    const long strideB = (long)4 * Nc;                         // advance 4 K per step

    for (int k0 = 0; k0 < K; k0 += 4) {
        v2f a = *(const v2f*)(Arow + k0);             // consecutive K pair
        // Preload all four B fragments (issues as one load clause).
        v2f b0, b1, b2, b3;
        b0.x = Bp[0];  b0.y = Bp[Nc + 0];
        b1.x = Bp[16]; b1.y = Bp[Nc + 16];
        b2.x = Bp[32]; b2.y = Bp[Nc + 32];
        b3.x = Bp[48]; b3.y = Bp[Nc + 48];
        // Four back-to-back WMMAs on independent accumulators.
        acc[0] = __builtin_amdgcn_wmma_f32_16x16x4_f32(
                     false, a, false, b0, (short)0, acc[0], false, false);
        acc[1] = __builtin_amdgcn_wmma_f32_16x16x4_f32(
                     false, a, false, b1, (short)0, acc[1], false, false);
        acc[2] = __builtin_amdgcn_wmma_f32_16x16x4_f32(
                     false, a, false, b2, (short)0, acc[2], false, false);
        acc[3] = __builtin_amdgcn_wmma_f32_16x16x4_f32(
                     false, a, false, b3, (short)0, acc[3], false, false);
        Bp += strideB;
    }

#pragma unroll
    for (int t = 0; t < 4; ++t)
#pragma unroll
        for (int j = 0; j < 8; ++j)
            C[(long)(m0 + j + 8 * half) * Nc + n0 + 16 * t + row] = acc[t][j];
}

// ---------------- bias (summed over relations) + leaky ReLU ----------------
__global__ void bias_lrelu_kernel(float* __restrict__ buf,
                                  const float* __restrict__ bias, // [R, cols]
                                  long total, int cols) {
    long i = (long)blockIdx.x * blockDim.x + threadIdx.x;
    if (i >= total) return;
    int col = (int)(i % cols);
    float bs = 0.0f;
#pragma unroll
    for (int r = 0; r < R_REL; ++r) bs += bias[r * cols + col];
    float v = buf[i] + bs;
    buf[i] = (v >= 0.0f) ? v : 0.01f * v;
}

// ---------------------------------------------------------------------------
extern "C" void kernel_launch(void* const* d_in, const int* in_sizes, int n_in,
                              void* d_out, int out_size, void* d_ws, size_t ws_size,
                              hipStream_t stream) {
    const float* x   = (const float*)d_in[0];
    const int*   src = (const int*)d_in[1];
    const int*   dst = (const int*)d_in[2];
    const float* W1  = (const float*)d_in[3];
    const float* b1  = (const float*)d_in[4];
    const float* W2  = (const float*)d_in[5];
    const float* b2  = (const float*)d_in[6];
    float* out = (float*)d_out;

    float* ws      = (float*)d_ws;
    float* dis_src = ws;                                  // R*N
    float* dis_dst = dis_src + (long)R_REL * N_NODES;     // R*N
    float* pre1    = dis_dst + (long)R_REL * N_NODES;     // N*HID (h1 in-place)
    float* agg     = pre1 + (long)N_NODES * HID;          // N*128 (agg / proj)

    // -- degrees (shared by both layers; layer1 uses r=6..8, layer2 all) ----
    hipMemsetAsync(dis_src, 0, (size_t)2 * R_REL * N_NODES * sizeof(float), stream);
    {
        int total = R_REL * E_EDGES;
        deg_count_kernel<<<(total + 255) / 256, 256, 0, stream>>>(src, dst, dis_src, dis_dst);
        int n = 2 * R_REL * N_NODES;
        deg_finalize_kernel<<<(n + 255) / 256, 256, 0, stream>>>(dis_src, n);
    }

    // -- layer 1: aggregate-first ------------------------------------------
    hipMemsetAsync(pre1, 0, (size_t)N_NODES * HID * sizeof(float), stream);
    const int edgeBlocks = (E_EDGES * 32 + 255) / 256;     // 1 wave per edge
    for (int r = 0; r < R_REL; ++r) {
        hipMemsetAsync(agg, 0, (size_t)N_NODES * F_IN * sizeof(float), stream);
        edge_scatter128_kernel<<<edgeBlocks, 256, 0, stream>>>(
            x, src + (long)r * E_EDGES, dst + (long)r * E_EDGES,
            dis_src + (long)r * N_NODES, dis_dst + (long)r * N_NODES,
            agg, E_EDGES, (r >= 6) ? 1 : 0);
        int tiles = (N_NODES / 16) * (HID / 64);           // 16x64 strips
        gemm_wmma_f32_kernel<<<(tiles + 7) / 8, 256, 0, stream>>>(
            agg, W1 + (long)r * F_IN * HID, pre1, N_NODES, HID, F_IN, 1);
    }
    {
        long total = (long)N_NODES * HID;
        bias_lrelu_kernel<<<(int)((total + 255) / 256), 256, 0, stream>>>(pre1, b1, total, HID);
    }

    // -- layer 2: weight-first ---------------------------------------------
    hipMemsetAsync(out, 0, (size_t)N_NODES * OUT_F * sizeof(float), stream);
    for (int r = 0; r < R_REL; ++r) {
        int tiles = (N_NODES / 16) * (OUT_F / 64);
        gemm_wmma_f32_kernel<<<(tiles + 7) / 8, 256, 0, stream>>>(
            pre1, W2 + (long)r * HID * OUT_F, agg, N_NODES, OUT_F, HID, 0);
        edge_scatter128_kernel<<<edgeBlocks, 256, 0, stream>>>(
            agg, src + (long)r * E_EDGES, dst + (long)r * E_EDGES,
            dis_src + (long)r * N_NODES, dis_dst + (long)r * N_NODES,
            out, E_EDGES, 1);
    }
    {
        long total = (long)N_NODES * OUT_F;
        bias_lrelu_kernel<<<(int)((total + 255) / 256), 256, 0, stream>>>(out, b2, total, OUT_F);
    }
}